// Capsule_fc_64836826300565
// MI455X (gfx1250) — compile-verified
//
#include <hip/hip_runtime.h>

// Capsule dynamic routing, fully fused, f32 WMMA (V_WMMA_F32_16X16X4_F32).
// Sizes fixed by reference: B=256, I=1152, K=8, J=10, D=16, R_ITERS=3.

#define R_ITERS 3
constexpr int B_ = 256, I_ = 1152, K_ = 8, J_ = 10, D_ = 16;
constexpr int BT = 16;        // batch tile = WMMA M
constexpr int NW = 8;         // waves per workgroup (wave32)
constexpr int NT = NW * 32;   // 256 threads

typedef __attribute__((ext_vector_type(2))) float v2f;
typedef __attribute__((ext_vector_type(8))) float v8f;

// LDS layout (floats):
//   bbL  [I_*BT]   logits b[i][b]        = 18432
//   sacc [NW*256]  per-wave s tiles      =  2048
//   sv   [256]     reduced s[b][d]
//   vv   [256]     squashed v[b][d]
//   red  [256]     reduction scratch
//   mxL  [16]      per-b row max
//   dnL  [16]      per-b softmax denom
constexpr int LDS_FLOATS = I_*BT + NW*256 + 256 + 256 + 256 + 16 + 16;

__global__ __launch_bounds__(NT)
void caps_routing_wmma(const float* __restrict__ x,   // [B,I,K]
                       const float* __restrict__ W,   // [I,J,D,K]
                       float* __restrict__ out) {     // [B,J,D]
  extern __shared__ float lds[];
  float* bbL  = lds;
  float* sacc = bbL + I_*BT;
  float* sv   = sacc + NW*256;
  float* vv   = sv + 256;
  float* red  = vv + 256;
  float* mxL  = red + 256;
  float* dnL  = mxL + 16;

  const int t    = threadIdx.x;
  const int lane = t & 31;
  const int wid  = t >> 5;
  const int lm   = lane & 15;   // WMMA: A row (batch) / B col (d)
  const int kh   = lane >> 4;   // WMMA: which K-pair this half-wave holds

  const int j  = blockIdx.x % J_;
  const int b0 = (blockIdx.x / J_) * BT;

  // init logits to zero
  for (int q = t; q < I_*BT; q += NT) bbL[q] = 0.0f;
  __syncthreads();

  // per-lane base pointers for the WMMA s-pass (A = x tile, B = W tile)
  // A layout: lane lm holds M=lm, K = 2*kh + {0,1}  -> float2 at x[b0+lm, i, 2kh]
  // B layout: lane lm holds N=lm, K-rows 2*kh + {0,1} -> float2 at W[i, j, lm, 2kh]
  const float* xA = x + (size_t)(b0 + lm) * I_ * K_ + 2*kh;
  const float* wB = W + (size_t)j * (D_*K_) + lm*K_ + 2*kh;

  for (int r = 1; r <= R_ITERS; ++r) {
    // ---- phase 1: row max of logits per batch b (softmax stabilization) ----
    {
      const int b = t & 15;
      float m = -3.402823466e38f;
      for (int i = (t >> 4); i < I_; i += 16) m = fmaxf(m, bbL[i*BT + b]);
      red[t] = m;
    }
    __syncthreads();
    if (t < 16) {
      float m = red[t];
      #pragma unroll
      for (int q = 1; q < 16; ++q) m = fmaxf(m, red[q*16 + t]);
      mxL[t] = m;
    }
    __syncthreads();

    // ---- phase 2: s[b,d] = sum_i exp(bb[i,b]-mx[b]) * u[b,i,d]  via WMMA ----
    // Softmax weight folded into the A operand; accumulate in the WMMA C/D.
    // Unrolled x2 with dual accumulators to break the D->C dependency chain.
    v8f acc0 = {}, acc1 = {};
    float dsum = 0.0f;
    const float mxb = mxL[lm];
    // I_/NW = 144 iterations per wave, even -> clean x2 unroll, no remainder.
    for (int i = wid; i < I_; i += 2*NW) {
      const int i2 = i + NW;
      float w0s = __expf(bbL[i *BT + lm] - mxb);
      float w1s = __expf(bbL[i2*BT + lm] - mxb);
      dsum += w0s + w1s;                            // softmax denominator
      const float* xp0 = xA + (size_t)i  * K_;
      const float* xp1 = xA + (size_t)i2 * K_;
      const float* wp0 = wB + (size_t)i  * (J_*D_*K_);
      const float* wp1 = wB + (size_t)i2 * (J_*D_*K_);
      v2f a00 = *(const v2f*)xp0;                   // i : K = 2kh, 2kh+1
      v2f a01 = *(const v2f*)(xp0 + 4);             // i : K = 4+2kh, 5+2kh
      v2f a10 = *(const v2f*)xp1;                   // i2
      v2f a11 = *(const v2f*)(xp1 + 4);
      v2f g00 = *(const v2f*)wp0;
      v2f g01 = *(const v2f*)(wp0 + 4);
      v2f g10 = *(const v2f*)wp1;
      v2f g11 = *(const v2f*)(wp1 + 4);
      a00 *= w0s; a01 *= w0s; a10 *= w1s; a11 *= w1s;
      acc0 = __builtin_amdgcn_wmma_f32_16x16x4_f32(false, a00, false, g00,
                                                   (short)0, acc0, false, false);
      acc1 = __builtin_amdgcn_wmma_f32_16x16x4_f32(false, a10, false, g10,
                                                   (short)0, acc1, false, false);
      acc0 = __builtin_amdgcn_wmma_f32_16x16x4_f32(false, a01, false, g01,
                                                   (short)0, acc0, false, false);
      acc1 = __builtin_amdgcn_wmma_f32_16x16x4_f32(false, a11, false, g11,
                                                   (short)0, acc1, false, false);
    }
    // stash per-wave tile: C/D element (b = rr + 8*kh, d = lm)
    #pragma unroll
    for (int rr = 0; rr < 8; ++rr)
      sacc[wid*256 + (rr + 8*kh)*16 + lm] = acc0[rr] + acc1[rr];
    if (lane < 16) red[wid*16 + lane] = dsum;       // halves are duplicates
    __syncthreads();

    // ---- phase 3: cross-wave reduce, normalize, squash ----
    if (t < 16) {
      float d = 0.0f;
      #pragma unroll
      for (int wq = 0; wq < NW; ++wq) d += red[wq*16 + t];
      dnL[t] = d;                                   // iter 1: exactly I_
    }
    {
      float s = 0.0f;
      #pragma unroll
      for (int wq = 0; wq < NW; ++wq) s += sacc[wq*256 + t];
      sv[t] = s;
    }
    __syncthreads();
    {
      const int b = t >> 4, d = t & 15;
      const float inv = 1.0f / dnL[b];
      float n2 = 0.0f;
      #pragma unroll
      for (int q = 0; q < 16; ++q) { float e = sv[b*16 + q] * inv; n2 += e*e; }
      const float sn = sv[b*16 + d] * inv;
      const float f  = __fsqrt_rn(n2) / (1.0f + n2);  // squash: s*|s|/(1+|s|^2)
      vv[t] = sn * f;
    }
    __syncthreads();

    if (r < R_ITERS) {
      // ---- phase 4: agreement  bb[i,b] += sum_d v[b,d]*u[b,i,d]
      // = sum_k x[b,i,k] * (sum_d v[b,d]*W[i,j,d,k])   (scalar per (i,b) owner)
      const int b = t & 15;
      float vr[16];
      #pragma unroll
      for (int d = 0; d < 16; ++d) vr[d] = vv[b*16 + d];
      for (int i = (t >> 4); i < I_; i += 16) {
        const float4* wp = (const float4*)(W + ((size_t)i*J_ + j) * (D_*K_));
        float4 t0 = {0,0,0,0}, t1 = {0,0,0,0};
        #pragma unroll
        for (int d = 0; d < 16; ++d) {
          float4 w0 = wp[2*d], w1 = wp[2*d + 1];
          t0.x += vr[d]*w0.x; t0.y += vr[d]*w0.y; t0.z += vr[d]*w0.z; t0.w += vr[d]*w0.w;
          t1.x += vr[d]*w1.x; t1.y += vr[d]*w1.y; t1.z += vr[d]*w1.z; t1.w += vr[d]*w1.w;
        }
        const float4* xp = (const float4*)(x + ((size_t)(b0 + b)*I_ + i) * K_);
        const float4 x0 = xp[0], x1 = xp[1];
        const float agree = x0.x*t0.x + x0.y*t0.y + x0.z*t0.z + x0.w*t0.w
                          + x1.x*t1.x + x1.y*t1.y + x1.z*t1.z + x1.w*t1.w;
        bbL[i*BT + b] += agree;   // unique owner thread per (i,b): no atomics
      }
      __syncthreads();
    } else {
      // ---- final: write v[b,j,d] ----
      const int b = t >> 4, d = t & 15;
      out[(size_t)(b0 + b) * (J_*D_) + (size_t)j * D_ + d] = vv[t];
    }
  }
}

extern "C" void kernel_launch(void* const* d_in, const int* in_sizes, int n_in,
                              void* d_out, int out_size, void* d_ws, size_t ws_size,
                              hipStream_t stream) {
  const float* x = (const float*)d_in[0];   // [256,1152,8]
  const float* W = (const float*)d_in[1];   // [1152,10,16,8]
  float* out = (float*)d_out;               // [256,10,16]
  const size_t shmem = (size_t)LDS_FLOATS * sizeof(float);  // 85,120 B
  dim3 grid((B_ / BT) * J_);                // 160 workgroups
  caps_routing_wmma<<<grid, dim3(NT), shmem, stream>>>(x, W, out);
}